// TreeLSTMModel_38070590112026
// MI455X (gfx1250) — compile-verified
//
#include <hip/hip_runtime.h>
#include <math.h>

// ---------------------------------------------------------------------------
// TreeLSTM forest on MI455X (gfx1250), fp32 WMMA (V_WMMA_F32_16X16X4_F32).
// Static perfect 4-ary forest: 150 trees x 341 nodes, level-order indexing.
//   depth ranges per tree: root 0 | 1..4 | 5..20 | 21..84 | leaves 85..340
//   children of local node q: 4q+1 .. 4q+4 (consecutive).
// ---------------------------------------------------------------------------

typedef __attribute__((ext_vector_type(2))) float v2f;
typedef __attribute__((ext_vector_type(8))) float v8f;

#define NTREES 150
#define NPT    341
#define NNODES (NTREES * NPT)   // 51150
#define HIDD   256
#define EMBD   256

// Native CDNA5 V_TANH_F32 (TRANS op): one instruction instead of a libm
// exp-based sequence.  Builtin preferred (compiler handles the TRANS-result
// hazard); asm fallback embeds the required V_NOP (ISA 7.4: 1 independent
// op or V_NOP after a TRANS before its output is consumed).
#if __has_builtin(__builtin_amdgcn_tanhf)
__device__ __forceinline__ float fast_tanh(float x) { return __builtin_amdgcn_tanhf(x); }
#else
__device__ __forceinline__ float fast_tanh(float x) {
  float r;
  asm("v_tanh_f32 %0, %1\n\tv_nop" : "=v"(r) : "v"(x));
  return r;
}
#endif
__device__ __forceinline__ float fast_sig(float x) {
  return fmaf(0.5f, fast_tanh(0.5f * x), 0.5f);  // sigmoid via tanh identity
}

__device__ __forceinline__ v8f wmma4(v2f a, v2f b, v8f c) {
  // D = A(16x4 f32) * B(4x16 f32) + C(16x16 f32), exact fp32, RNE.
  return __builtin_amdgcn_wmma_f32_16x16x4_f32(false, a, false, b, (short)0, c,
                                               false, false);
}

// ---------------------------------------------------------------------------
// Kernel 1: leaf pass.  Fuses embedding gather + subtoken mean into the LDS
// A-tile build, then iou0 = embeds @ W_iou (three 16-wide column groups per
// wave: i / o / u thirds), epilogue computes leaf c,h.
// grid.x = 38400/16 = 2400 tiles of 16 leaf rows; block = 512 (16 waves).
// ---------------------------------------------------------------------------
__global__ __launch_bounds__(512) void leaf_iou_kernel(
    const int* __restrict__ feat, const float* __restrict__ emb,
    const float* __restrict__ W_iou, const float* __restrict__ b_iou,
    float* __restrict__ h, float* __restrict__ c) {
  __shared__ float As[16][260];  // pad 256->260: conflict-free lane reads
  const int tid = threadIdx.x;
  const int mbase = blockIdx.x * 16;  // global leaf-row base (always full tile)

  // Build A tile: embeds = mean over 8 subtokens (tables are L2-resident).
  for (int e = 0; e < 8; ++e) {
    int idx = tid + e * 512;
    int r = idx >> 8, k = idx & 255;
    int lr = mbase + r;
    int tree = lr >> 8;                       // 256 leaves per tree
    int node = tree * NPT + 85 + (lr & 255);  // leaf node index
    const int* f8 = feat + node * 8;
    float s = 0.0f;
#pragma unroll
    for (int t = 0; t < 8; ++t) s += emb[f8[t] * EMBD + k];
    As[r][k] = s * 0.125f;
  }
  __syncthreads();

  const int lane = tid & 31, w = tid >> 5;
  const int half = lane >> 4, l15 = lane & 15;
  const int cb = w * 16;  // column group within each 256-wide third
  v8f ai = {}, ao = {}, au = {};
#pragma unroll 4
  for (int k0 = 0; k0 < 256; k0 += 4) {
    int ka = k0 + 2 * half;
    v2f a;
    a.x = As[l15][ka];
    a.y = As[l15][ka + 1];
    const float* w0 = W_iou + ka * 768;
    v2f bi, bo, bu;
    bi.x = w0[cb + l15];        bi.y = w0[768 + cb + l15];
    bo.x = w0[256 + cb + l15];  bo.y = w0[768 + 256 + cb + l15];
    bu.x = w0[512 + cb + l15];  bu.y = w0[768 + 512 + cb + l15];
    ai = wmma4(a, bi, ai);
    ao = wmma4(a, bo, ao);
    au = wmma4(a, bu, au);
  }
  const int col = cb + l15;
  const float bii = b_iou[col], bio = b_iou[256 + col], biu = b_iou[512 + col];
#pragma unroll
  for (int v = 0; v < 8; ++v) {
    int lr = mbase + v + 8 * half;  // C layout: VGPR v -> row v (+8 hi half)
    int tree = lr >> 8;
    int node = tree * NPT + 85 + (lr & 255);
    float iv = ai[v] + bii, ov = ao[v] + bio, uv = au[v] + biu;
    float cv = fast_sig(iv) * fast_tanh(uv);
    float hv = fast_sig(ov) * fast_tanh(cv);
    c[node * HIDD + col] = cv;
    h[node * HIDD + col] = hv;
  }
}

// ---------------------------------------------------------------------------
// Kernel 2: per-edge forget gate.  A = h[children] (contiguous range per
// tree), GEMM vs Uf_W, epilogue: f*c_child, then reduce the 4-child groups
// inside the C fragment (rows 0..3 / 4..7 of each half) -> parent c_sum.
// clen = 1<<cshift (power of 4), so tree/local split is shift+mask.
// ---------------------------------------------------------------------------
__global__ __launch_bounds__(512) void fgate_kernel(
    const float* __restrict__ h, const float* __restrict__ c,
    const float* __restrict__ UfW, const float* __restrict__ Ufb,
    float* __restrict__ csum, int cstart, int cshift, int rows) {
  __shared__ float As[16][260];
  const int tid = threadIdx.x;
  const int mbase = blockIdx.x * 16;
  const int cmask = (1 << cshift) - 1;
  for (int e = 0; e < 8; ++e) {
    int idx = tid + e * 512;
    int r = idx >> 8, k = idx & 255;
    int cr = mbase + r;
    if (cr >= rows) cr = rows - 1;  // clamp partial tile (masked at store)
    int node = (cr >> cshift) * NPT + cstart + (cr & cmask);
    As[r][k] = h[node * HIDD + k];
  }
  __syncthreads();

  const int lane = tid & 31, w = tid >> 5;
  const int half = lane >> 4, l15 = lane & 15;
  const int cb = w * 16;
  const int col = cb + l15;

  // Pre-resolve the 8 scattered c[child] operands for the epilogue and
  // prefetch them so the lines land under the WMMA loop.
  const float* cp[8];
#pragma unroll
  for (int v = 0; v < 8; ++v) {
    int cr = mbase + v + 8 * half;
    int crc = cr < rows ? cr : rows - 1;
    int node = (crc >> cshift) * NPT + cstart + (crc & cmask);
    cp[v] = c + node * HIDD + col;
    __builtin_prefetch(cp[v], 0, 0);  // -> global_prefetch_b8
  }

  v8f acc = {};
#pragma unroll 4
  for (int k0 = 0; k0 < 256; k0 += 4) {
    int ka = k0 + 2 * half;
    v2f a;
    a.x = As[l15][ka];
    a.y = As[l15][ka + 1];
    v2f b;
    b.x = UfW[ka * HIDD + col];
    b.y = UfW[(ka + 1) * HIDD + col];
    acc = wmma4(a, b, acc);
  }
  const float bb = Ufb[col];
  const int P = rows >> 2;  // 4 children per parent
  float s0 = 0.0f, s1 = 0.0f;
#pragma unroll
  for (int v = 0; v < 8; ++v) {
    float fc = fast_sig(acc[v] + bb) * (*cp[v]);
    if (v < 4) s0 += fc; else s1 += fc;
  }
  int p0 = blockIdx.x * 4 + 2 * half;  // parent rows covered by this half
  if (p0 < P) csum[p0 * HIDD + col] = s0;
  if (p0 + 1 < P) csum[(p0 + 1) * HIDD + col] = s1;
}

// ---------------------------------------------------------------------------
// Kernel 3: parent update.  A-tile = h_tild (sum of 4 child h rows, built on
// the fly), GEMM vs U_iou (3 column groups), epilogue applies gates + c_sum.
// plen = 1<<pshift.
// ---------------------------------------------------------------------------
__global__ __launch_bounds__(512) void iou_kernel(
    float* __restrict__ h, float* __restrict__ c,
    const float* __restrict__ Uiou, const float* __restrict__ biou,
    const float* __restrict__ csum, int pstart, int pshift, int P) {
  __shared__ float As[16][260];
  const int tid = threadIdx.x;
  const int mbase = blockIdx.x * 16;
  const int pmask = (1 << pshift) - 1;
  for (int e = 0; e < 8; ++e) {
    int idx = tid + e * 512;
    int r = idx >> 8, k = idx & 255;
    int pr = mbase + r;
    if (pr >= P) pr = P - 1;
    int tree = pr >> pshift;
    int q = pstart + (pr & pmask);                   // parent local node
    int base = (tree * NPT + 4 * q + 1) * HIDD + k;  // 4 consecutive children
    As[r][k] = h[base] + h[base + HIDD] + h[base + 2 * HIDD] + h[base + 3 * HIDD];
  }
  __syncthreads();

  const int lane = tid & 31, w = tid >> 5;
  const int half = lane >> 4, l15 = lane & 15;
  const int cb = w * 16;
  const int col = cb + l15;

  // Prefetch the csum rows consumed in the epilogue.
#pragma unroll
  for (int v = 0; v < 8; ++v) {
    int pr = mbase + v + 8 * half;
    if (pr < P) __builtin_prefetch(csum + pr * HIDD + col, 0, 0);
  }

  v8f ai = {}, ao = {}, au = {};
#pragma unroll 4
  for (int k0 = 0; k0 < 256; k0 += 4) {
    int ka = k0 + 2 * half;
    v2f a;
    a.x = As[l15][ka];
    a.y = As[l15][ka + 1];
    const float* w0 = Uiou + ka * 768;
    v2f bi, bo, bu;
    bi.x = w0[cb + l15];        bi.y = w0[768 + cb + l15];
    bo.x = w0[256 + cb + l15];  bo.y = w0[768 + 256 + cb + l15];
    bu.x = w0[512 + cb + l15];  bu.y = w0[768 + 512 + cb + l15];
    ai = wmma4(a, bi, ai);
    ao = wmma4(a, bo, ao);
    au = wmma4(a, bu, au);
  }
  const float bii = biou[col], bio = biou[256 + col], biu = biou[512 + col];
#pragma unroll
  for (int v = 0; v < 8; ++v) {
    int pr = mbase + v + 8 * half;
    if (pr < P) {
      int node = (pr >> pshift) * NPT + pstart + (pr & pmask);
      float iv = ai[v] + bii, ov = ao[v] + bio, uv = au[v] + biu;
      float cv = fast_sig(iv) * fast_tanh(uv) + csum[pr * HIDD + col];
      float hv = fast_sig(ov) * fast_tanh(cv);
      c[node * HIDD + col] = cv;
      h[node * HIDD + col] = hv;
    }
  }
}

// Mean graph readout + ReLU.  grid = 150 trees, block = 256 columns.
__global__ void readout_kernel(const float* __restrict__ h, float* __restrict__ m) {
  const int tree = blockIdx.x, col = threadIdx.x;
  const float* hp = h + tree * NPT * HIDD + col;
  float s = 0.0f;
  for (int j = 0; j < NPT; ++j) s += hp[j * HIDD];
  s *= (1.0f / (float)NPT);
  m[tree * HIDD + col] = s > 0.0f ? s : 0.0f;
}

// Head: [m1 m2] @ Wf + bf, leaky-relu(0.01), softmax(2).  Tiny: 1 thread/tree.
__global__ void head_kernel(const float* __restrict__ m1, const float* __restrict__ m2,
                            const float* __restrict__ Wf, const float* __restrict__ bf,
                            float* __restrict__ out) {
  int t = blockIdx.x * blockDim.x + threadIdx.x;
  if (t >= NTREES) return;
  float s0 = bf[0], s1 = bf[1];
  for (int j = 0; j < HIDD; ++j) {
    float a = m1[t * HIDD + j], b = m2[t * HIDD + j];
    s0 += a * Wf[j * 2 + 0] + b * Wf[(HIDD + j) * 2 + 0];
    s1 += a * Wf[j * 2 + 1] + b * Wf[(HIDD + j) * 2 + 1];
  }
  s0 = s0 > 0.0f ? s0 : 0.01f * s0;
  s1 = s1 > 0.0f ? s1 : 0.01f * s1;
  float mx = fmaxf(s0, s1);
  float e0 = __expf(s0 - mx), e1 = __expf(s1 - mx);
  float inv = 1.0f / (e0 + e1);
  out[t * 2 + 0] = e0 * inv;
  out[t * 2 + 1] = e1 * inv;
}

// ---------------------------------------------------------------------------
extern "C" void kernel_launch(void* const* d_in, const int* in_sizes, int n_in,
                              void* d_out, int out_size, void* d_ws, size_t ws_size,
                              hipStream_t stream) {
  (void)in_sizes; (void)n_in; (void)out_size; (void)ws_size;
  // Workspace layout (floats): h[N*256] | c[N*256] | csum[9600*256] | m1 | m2
  float* ws = (float*)d_ws;
  float* h  = ws;
  float* c  = h + (size_t)NNODES * HIDD;
  float* cs = c + (size_t)NNODES * HIDD;
  float* m1 = cs + (size_t)9600 * HIDD;
  float* m2 = m1 + (size_t)NTREES * HIDD;

  // Level tables: children range (cstart, clen=1<<cshift) feeding parents
  // (pstart, plen=1<<pshift).
  const int cstart[4] = {85, 21, 5, 1};
  const int cshift[4] = {8, 6, 4, 2};   // 256, 64, 16, 4 children per tree
  const int pstart[4] = {21, 5, 1, 0};
  const int pshift[4] = {6, 4, 2, 0};   // 64, 16, 4, 1 parents per tree

  for (int b = 0; b < 2; ++b) {
    const int*   feat = (const int*)d_in[0 + b];
    const float* emb  = (const float*)d_in[2 + b];
    const float* Wiou = (const float*)d_in[4 + b];
    const float* Uiou = (const float*)d_in[6 + b];
    const float* UfW  = (const float*)d_in[8 + 2 * b];
    const float* Ufb  = (const float*)d_in[9 + 2 * b];
    const float* biou = (const float*)d_in[12 + b];
    float* m = b ? m2 : m1;

    leaf_iou_kernel<<<2400, 512, 0, stream>>>(feat, emb, Wiou, biou, h, c);
    for (int l = 0; l < 4; ++l) {
      int rows = NTREES << cshift[l];  // active child edges this level
      int P    = NTREES << pshift[l];  // parents this level
      fgate_kernel<<<(rows + 15) / 16, 512, 0, stream>>>(h, c, UfW, Ufb, cs,
                                                         cstart[l], cshift[l], rows);
      iou_kernel<<<(P + 15) / 16, 512, 0, stream>>>(h, c, Uiou, biou, cs,
                                                    pstart[l], pshift[l], P);
    }
    readout_kernel<<<NTREES, 256, 0, stream>>>(h, m);
  }
  head_kernel<<<1, 256, 0, stream>>>(m1, m2, (const float*)d_in[14],
                                     (const float*)d_in[15], (float*)d_out);
}